// EdgeTrans_38482906972444
// MI455X (gfx1250) — compile-verified
//
#include <hip/hip_runtime.h>
#include <math.h>

#define N_NODES 2048
#define N_EDGES 65536
#define FDIM    256
#define ROWLEN  2048      // per-graph map row length
#define CATLEN  4096      // concatenated softmax row length

typedef __attribute__((ext_vector_type(2))) float v2f;
typedef __attribute__((ext_vector_type(4))) float v4f;
typedef __attribute__((ext_vector_type(8))) float v8f;

#if defined(__has_builtin)
#if __has_builtin(__builtin_amdgcn_wmma_f32_16x16x4_f32)
#define HAVE_WMMA_F32X4 1
#endif
#endif

// ---------------- CSR build kernels ----------------

__global__ void zero_kernel(int* __restrict__ p, int n) {
    int i = blockIdx.x * blockDim.x + threadIdx.x;
    if (i < n) p[i] = 0;
}

__global__ void count_kernel(const int* __restrict__ dst, int* __restrict__ deg, int e) {
    int i = blockIdx.x * blockDim.x + threadIdx.x;
    if (i < e) atomicAdd(&deg[dst[i]], 1);
}

// single block, 256 threads, exclusive scan over 2048 counts -> row_ptr[2049]
__global__ void scan_kernel(const int* __restrict__ deg, int* __restrict__ rp) {
    __shared__ int part[256];
    int t = threadIdx.x;
    int base = t * 8;
    int local[8];
    int s = 0;
    for (int i = 0; i < 8; ++i) { local[i] = s; s += deg[base + i]; }
    part[t] = s;
    __syncthreads();
    if (t == 0) {
        int r = 0;
        for (int i = 0; i < 256; ++i) { int v = part[i]; part[i] = r; r += v; }
        rp[N_NODES] = r;
    }
    __syncthreads();
    int off = part[t];
    for (int i = 0; i < 8; ++i) rp[base + i] = off + local[i];
}

__global__ void scatter_kernel(const int* __restrict__ dst, const int* __restrict__ rp,
                               int* __restrict__ cur, int* __restrict__ eid, int e) {
    int i = blockIdx.x * blockDim.x + threadIdx.x;
    if (i < e) {
        int d = dst[i];
        int pos = atomicAdd(&cur[d], 1);
        eid[rp[d] + pos] = i;
    }
}

// canonicalize each node's edge list (ascending edge id) -> deterministic fp sums
__global__ void sort_kernel(const int* __restrict__ rp, int* __restrict__ eid) {
    int n = blockIdx.x * blockDim.x + threadIdx.x;
    if (n < N_NODES) {
        int b = rp[n], e = rp[n + 1];
        for (int i = b + 1; i < e; ++i) {
            int v = eid[i];
            int j = i - 1;
            while (j >= b && eid[j] > v) { eid[j + 1] = eid[j]; --j; }
            eid[j + 1] = v;
        }
    }
}

// ---------------- fused propagate + softmax + WMMA einsum ----------------
// grid: N_NODES/16 blocks; block: 512 threads = 16 waves (wave32).
// wave w owns node blockIdx.x*16+w for propagate/softmax, then computes
// output columns [16w, 16w+16) of the 16x256 output tile via WMMA f32 16x16x4.

__global__ __launch_bounds__(512)
void fused_kernel(const float* __restrict__ img_map_pre,
                  const float* __restrict__ kg_map_pre,
                  const float* __restrict__ img_edge_map,
                  const float* __restrict__ kg_edge_map,
                  const float* __restrict__ img_feat,
                  const float* __restrict__ kg_feat,
                  const int*   __restrict__ img_src,
                  const int*   __restrict__ kg_src,
                  const int*   __restrict__ rp_img, const int* __restrict__ eid_img,
                  const int*   __restrict__ rp_kg,  const int* __restrict__ eid_kg,
                  float*       __restrict__ out)
{
    __shared__ float rows[16][CATLEN];   // 256 KB (<= 320 KB WGP LDS)

    const int lane = threadIdx.x & 31;
    const int w    = threadIdx.x >> 5;   // wave id 0..15
    const int node = blockIdx.x * 16 + w;

    // ---- phase 1: propagate both graphs, row parked in LDS ----
    for (int g = 0; g < 2; ++g) {
        const float* mp  = g ? kg_map_pre  : img_map_pre;
        const float* em  = g ? kg_edge_map : img_edge_map;
        const int* srcA  = g ? kg_src  : img_src;
        const int* rp    = g ? rp_kg   : rp_img;
        const int* eid   = g ? eid_kg  : eid_img;

        int beg = rp[node];
        int cnt = rp[node + 1] - beg;

        v4f acc[16];
        if (cnt == 0) {
            // no in-edges: keep map_pre (reference's `continue`)
            const v4f* r4 = (const v4f*)(mp + (size_t)node * ROWLEN);
            #pragma unroll
            for (int j = 0; j < 16; ++j) acc[j] = r4[j * 32 + lane];
        } else {
            #pragma unroll
            for (int j = 0; j < 16; ++j) acc[j] = (v4f){0.f, 0.f, 0.f, 0.f};
            for (int idx = 0; idx < cnt; ++idx) {
                int e = eid[beg + idx];
                int s = srcA[e];
                if (idx + 1 < cnt) {
                    // prefetch next edge's rows; lanes 256B apart cover the 8KB row
                    int en = eid[beg + idx + 1];
                    int sn = srcA[en];
                    __builtin_prefetch(em + (size_t)en * ROWLEN + lane * 64, 0, 0);
                    __builtin_prefetch(mp + (size_t)sn * ROWLEN + lane * 64, 0, 0);
                }
                const v4f* u4 = (const v4f*)(mp + (size_t)s * ROWLEN);
                const v4f* e4 = (const v4f*)(em + (size_t)e * ROWLEN);
                #pragma unroll
                for (int j = 0; j < 16; ++j) {
                    v4f a = u4[j * 32 + lane];
                    // edge_map is streamed once (1 GB): non-temporal, don't evict
                    // the L2-resident map_pre / feats working set
                    v4f b = __builtin_nontemporal_load(&e4[j * 32 + lane]);
                    acc[j] += a * b;
                }
            }
        }
        v4f* d4 = (v4f*)&rows[w][g * ROWLEN];
        #pragma unroll
        for (int j = 0; j < 16; ++j) d4[j * 32 + lane] = acc[j];
    }

    // ---- phase 2: softmax over the wave's own 4096-wide row ----
    float m = -3.402823466e38f;
    for (int j = lane; j < CATLEN; j += 32) m = fmaxf(m, rows[w][j]);
    #pragma unroll
    for (int o = 16; o; o >>= 1) m = fmaxf(m, __shfl_xor(m, o, 32));

    float ssum = 0.f;
    for (int j = lane; j < CATLEN; j += 32) {
        float e = __expf(rows[w][j] - m);
        rows[w][j] = e;
        ssum += e;
    }
    #pragma unroll
    for (int o = 16; o; o >>= 1) ssum += __shfl_xor(ssum, o, 32);
    float inv = 1.0f / ssum;
    for (int j = lane; j < CATLEN; j += 32) rows[w][j] *= inv;

    __syncthreads();   // all 16 rows normalized; every wave now reads all rows

    // ---- phase 3: [16 x 4096] @ [4096 x 256] via V_WMMA_F32_16X16X4_F32 ----
    // K loop split into img half / kg half: branch-free, B pointer is a pure
    // induction (+4*FDIM per step).
    const int half  = lane >> 4;   // 0: lanes 0-15, 1: lanes 16-31
    const int l16   = lane & 15;
    const int fbase = w * 16;      // this wave's 16 output columns
    const int k0off = half * 2;    // A/B K-lane offset within each 4-wide step

#if defined(HAVE_WMMA_F32X4)
    v8f c = {};
    {   // K in [0, 2048): B rows from img_feat
        const float* bp = img_feat + (size_t)k0off * FDIM + fbase + l16;
        #pragma unroll 8
        for (int kb = 0; kb < ROWLEN; kb += 4) {
            v2f a;
            a.x = rows[l16][kb + k0off];
            a.y = rows[l16][kb + k0off + 1];
            v2f b;
            b.x = bp[0];
            b.y = bp[FDIM];
            c = __builtin_amdgcn_wmma_f32_16x16x4_f32(false, a, false, b, (short)0, c,
                                                      false, false);
            bp += 4 * FDIM;
        }
    }
    {   // K in [2048, 4096): B rows from kg_feat
        const float* bp = kg_feat + (size_t)k0off * FDIM + fbase + l16;
        #pragma unroll 8
        for (int kb = 0; kb < ROWLEN; kb += 4) {
            v2f a;
            a.x = rows[l16][ROWLEN + kb + k0off];
            a.y = rows[l16][ROWLEN + kb + k0off + 1];
            v2f b;
            b.x = bp[0];
            b.y = bp[FDIM];
            c = __builtin_amdgcn_wmma_f32_16x16x4_f32(false, a, false, b, (short)0, c,
                                                      false, false);
            bp += 4 * FDIM;
        }
    }
    // C/D layout: VGPR r -> M=r (lanes 0-15, N=lane) / M=8+r (lanes 16-31, N=lane-16)
    const int mbase = half * 8;
    #pragma unroll
    for (int r = 0; r < 8; ++r) {
        int nm = blockIdx.x * 16 + mbase + r;
        out[(size_t)nm * FDIM + fbase + l16] = c[r];
    }
#else
    // scalar fallback (compile-safe; signals wmma=0 in histogram)
    for (int mrow = half; mrow < 16; mrow += 2) {
        float accv = 0.f;
        for (int k = 0; k < CATLEN; ++k) {
            const float* fr = (k < ROWLEN) ? (img_feat + (size_t)k * FDIM)
                                           : (kg_feat + (size_t)(k - ROWLEN) * FDIM);
            accv = fmaf(rows[mrow][k], fr[fbase + l16], accv);
        }
        out[(size_t)(blockIdx.x * 16 + mrow) * FDIM + fbase + l16] = accv;
    }
#endif
}

// ---------------- host side ----------------

extern "C" void kernel_launch(void* const* d_in, const int* in_sizes, int n_in,
                              void* d_out, int out_size, void* d_ws, size_t ws_size,
                              hipStream_t stream) {
    const float* img_map_pre  = (const float*)d_in[0];
    const float* kg_map_pre   = (const float*)d_in[1];
    const float* img_edge_map = (const float*)d_in[2];
    const float* kg_edge_map  = (const float*)d_in[3];
    const float* img_feat     = (const float*)d_in[4];
    const float* kg_feat      = (const float*)d_in[5];
    const int*   img_src      = (const int*)d_in[6];
    const int*   img_dst      = (const int*)d_in[7];
    const int*   kg_src       = (const int*)d_in[8];
    const int*   kg_dst       = (const int*)d_in[9];
    float* out = (float*)d_out;

    // workspace layout (ints): deg_img, deg_kg, cur_img, cur_kg must be adjacent
    int* wp = (int*)d_ws;
    int* deg_img = wp;  wp += N_NODES;
    int* deg_kg  = wp;  wp += N_NODES;
    int* cur_img = wp;  wp += N_NODES;
    int* cur_kg  = wp;  wp += N_NODES;
    int* rp_img  = wp;  wp += N_NODES + 1;
    int* rp_kg   = wp;  wp += N_NODES + 1;
    int* eid_img = wp;  wp += N_EDGES;
    int* eid_kg  = wp;  wp += N_EDGES;

    zero_kernel<<<(4 * N_NODES + 255) / 256, 256, 0, stream>>>((int*)d_ws, 4 * N_NODES);

    count_kernel<<<N_EDGES / 256, 256, 0, stream>>>(img_dst, deg_img, N_EDGES);
    count_kernel<<<N_EDGES / 256, 256, 0, stream>>>(kg_dst,  deg_kg,  N_EDGES);

    scan_kernel<<<1, 256, 0, stream>>>(deg_img, rp_img);
    scan_kernel<<<1, 256, 0, stream>>>(deg_kg,  rp_kg);

    scatter_kernel<<<N_EDGES / 256, 256, 0, stream>>>(img_dst, rp_img, cur_img, eid_img, N_EDGES);
    scatter_kernel<<<N_EDGES / 256, 256, 0, stream>>>(kg_dst,  rp_kg,  cur_kg,  eid_kg,  N_EDGES);

    sort_kernel<<<N_NODES / 256, 256, 0, stream>>>(rp_img, eid_img);
    sort_kernel<<<N_NODES / 256, 256, 0, stream>>>(rp_kg,  eid_kg);

    fused_kernel<<<N_NODES / 16, 512, 0, stream>>>(
        img_map_pre, kg_map_pre, img_edge_map, kg_edge_map,
        img_feat, kg_feat, img_src, kg_src,
        rp_img, eid_img, rp_kg, eid_kg, out);
}